// Model_45140106281625
// MI455X (gfx1250) — compile-verified
//
#include <hip/hip_runtime.h>
#include <math.h>

// ---------------- problem constants ----------------
#define N_ATOMS   32
#define N_EDGES   448
#define CCH       32
#define NDOF      96                  // 32 atoms * 3
#define NPAIRS    ((NDOF * (NDOF + 1)) / 2)   // 4656 upper-tri pairs incl diag
#define CUTOFF_R  5.0f
#define INV_AVG   (1.0f / 14.26f)
#define FD_H      0.02f
#define PI_F      3.14159265358979f

// ---------------- LDS layout (floats, dynamic) ----------------
// pos:96  r:448  u:1344  wpath:4032  s0/s1:1024x2  v0/v1:3072x2  T0/T1:6144x2
// agg:3072  wcomb:4096  gv:1024  gT:1024  red:64
#define OFF_POS    0
#define OFF_R      96
#define OFF_U      544
#define OFF_WPATH  1888
#define OFF_S0     5920
#define OFF_S1     6944
#define OFF_V0     7968
#define OFF_V1     11040
#define OFF_T0     14112
#define OFF_T1     20256
#define OFF_AGG    26400
#define OFF_WC     29472
#define OFF_GV     33568
#define OFF_GT     34592
#define OFF_RED    35616
#define SMEM_FLOATS 35680   // 142,720 bytes dynamic LDS

typedef __attribute__((ext_vector_type(2))) float v2f;
typedef __attribute__((ext_vector_type(8))) float v8f;

struct P {
    const float* positions;  // [32,3]
    const float* Wr1;        // [3,8,64]
    const float* Wr2;        // [3,64,9]
    const float* W0_s;       // [11,32]
    const float* W0_v;       // [11,32]
    const float* W0_T;       // [11,32]
    const float* Wsk0_s;     // [9,32]
    const float* Wsk0_v;     // [1,32]
    const float* Wsk0_T;     // [1,32]
    const float* W_s;        // [2,96,32]
    const float* W_v;        // [2,96,32]
    const float* W_T;        // [2,96,32]
    const float* Wsk;        // [2,3,32,32]
    const float* Wg;         // [3,2,32,32]
    const float* Wro1;       // [32,8]
    const float* Wro2;       // [8,1]
    const int*   species;    // [32]
    const int*   senders;    // [448]
    const int*   receivers;  // [448]
    float*       out;        // [96,96]
};

__device__ __forceinline__ float sigmoidf_(float x) { return 1.0f / (1.0f + expf(-x)); }

// ---------------------------------------------------------------------------
// Stage combined mixing operand Wcomb[128][32] into LDS.
// Rows 0..95  : message-mix weights (scaled by 1/AVG_NEIGH), channel blocks
//               [path0: 0..31][path1: 32..63][path2: 64..95].
// Rows 96..127: self-interaction skip weights.
// Layer 0 uses zero-padded placement of the (11-row / 9-row / 1-row) weights.
// ---------------------------------------------------------------------------
__device__ void stage_wcomb(float* W, int layer, int irrep, const P& p)
{
    const int tid = threadIdx.x;
    if (layer == 0) {
        const float* W0 = (irrep == 0) ? p.W0_s : (irrep == 1) ? p.W0_v : p.W0_T;
        for (int idx = tid; idx < 128 * 32; idx += 256) W[idx] = 0.0f;
        __syncthreads();
        for (int idx = tid; idx < 11 * 32; idx += 256) {
            int r = idx >> 5, c = idx & 31;
            int dst = (r < 9) ? r : ((r == 9) ? 32 : 64);   // s-ch r | v-ch0 | T-ch0
            W[dst * 32 + c] = W0[idx] * INV_AVG;
        }
        if (irrep == 0) {
            for (int idx = tid; idx < 9 * 32; idx += 256) {
                int r = idx >> 5, c = idx & 31;
                W[(96 + r) * 32 + c] = p.Wsk0_s[idx];
            }
        } else {
            const float* Wk = (irrep == 1) ? p.Wsk0_v : p.Wsk0_T;
            for (int idx = tid; idx < 32; idx += 256) W[96 * 32 + idx] = Wk[idx];
        }
    } else {
        const float* Wm = (((irrep == 0) ? p.W_s : (irrep == 1) ? p.W_v : p.W_T)
                           + (layer - 1) * 96 * 32);
        for (int idx = tid; idx < 96 * 32; idx += 256) W[idx] = Wm[idx] * INV_AVG;
        const float* Wk = p.Wsk + ((layer - 1) * 3 + irrep) * 1024;
        for (int idx = tid; idx < 1024; idx += 256) W[3072 + idx] = Wk[idx];
    }
}

// ---------------------------------------------------------------------------
// D[32,32] = [agg(32x96) | feat(32x32)] (32x128) @ Wcomb(128x32)
// via V_WMMA_F32_16X16X4_F32.  2x2 output tiles of 16x16, K = 128 (32 steps).
// feat/out strided by fs with component offset fo (1 for s, 3 for v, 6 for T).
// ---------------------------------------------------------------------------
__device__ void mix_wmma(const float* __restrict__ agg,
                         const float* __restrict__ feat, int fs, int fo,
                         const float* __restrict__ Wc,
                         float* __restrict__ outp)
{
    const int lane = threadIdx.x & 31;
    const int wave = threadIdx.x >> 5;
    const int lh = lane >> 4;       // half-wave: K sub-pair / row offset select
    const int ll = lane & 15;
    for (int job = wave; job < 4; job += 8) {       // wave-uniform -> EXEC all 1s
        const int tm = job >> 1, tn = job & 1;
        v8f acc = {};
        const int m = tm * 16 + ll;
        const int n = tn * 16 + ll;
        for (int kk = 0; kk < 32; ++kk) {
            const int k0 = kk * 4 + lh * 2;
            const int k1 = k0 + 1;
            v2f a, b;
            a.x = (k0 < 96) ? agg[m * 96 + k0] : feat[(m * 32 + (k0 - 96)) * fs + fo];
            a.y = (k1 < 96) ? agg[m * 96 + k1] : feat[(m * 32 + (k1 - 96)) * fs + fo];
            b.x = Wc[k0 * 32 + n];
            b.y = Wc[k1 * 32 + n];
            acc = __builtin_amdgcn_wmma_f32_16x16x4_f32(
                false, a, false, b, (short)0, acc, false, false);
        }
#pragma unroll
        for (int r = 0; r < 8; ++r) {
            const int mr = tm * 16 + r + lh * 8;    // C/D layout: VGPR r -> M=r / r+8
            outp[(mr * 32 + n) * fs + fo] = acc[r];
        }
    }
}

// Gate matmuls: gv = s_o @ Wg[l][0], gT = s_o @ Wg[l][1].  8 tiles -> 8 waves.
__device__ void gates_wmma(const float* __restrict__ s_o,
                           const float* __restrict__ WgL,
                           float* __restrict__ gv, float* __restrict__ gT)
{
    const int lane = threadIdx.x & 31;
    const int wave = threadIdx.x >> 5;
    const int lh = lane >> 4, ll = lane & 15;
    const int g = wave >> 2, t = wave & 3;
    const int tm = t >> 1, tn = t & 1;
    const float* B = WgL + g * 32 * 32;
    float* outp = g ? gT : gv;
    v8f acc = {};
    const int m = tm * 16 + ll;
    const int n = tn * 16 + ll;
    for (int kk = 0; kk < 8; ++kk) {
        const int k0 = kk * 4 + lh * 2;
        v2f a, b;
        a.x = s_o[m * 32 + k0];
        a.y = s_o[m * 32 + k0 + 1];
        b.x = B[k0 * 32 + n];
        b.y = B[(k0 + 1) * 32 + n];
        acc = __builtin_amdgcn_wmma_f32_16x16x4_f32(
            false, a, false, b, (short)0, acc, false, false);
    }
#pragma unroll
    for (int r = 0; r < 8; ++r)
        outp[(tm * 16 + r + lh * 8) * 32 + n] = acc[r];
}

// ---------------------------------------------------------------------------
// Edge scatter passes: message construction + LDS-atomic aggregation.
// Symmetric tensors stored as 6 comps: 0:xx 1:yy 2:zz 3:xy 4:xz 5:yz.
// ---------------------------------------------------------------------------
__device__ void edge_pass_S(const P& p, float* agg, const float* u, const float* wp,
                            const float* si, const float* vi, const float* Ti)
{
    for (int e = threadIdx.x; e < N_EDGES; e += 256) {
        const int sn = p.senders[e], rn = p.receivers[e];
        const float ux = u[e * 3], uy = u[e * 3 + 1], uz = u[e * 3 + 2];
        const float uu0 = ux * ux - (1.f / 3.f), uu1 = uy * uy - (1.f / 3.f),
                    uu2 = uz * uz - (1.f / 3.f), uu3 = ux * uy, uu4 = ux * uz,
                    uu5 = uy * uz;
        const float w0 = wp[e * 9], w1 = wp[e * 9 + 1], w2 = wp[e * 9 + 2];
        for (int c = 0; c < CCH; ++c) {
            const float ss = si[sn * 32 + c];
            const float* vv = vi + (sn * 32 + c) * 3;
            const float dv = vv[0] * ux + vv[1] * uy + vv[2] * uz;
            const float* TT = Ti + (sn * 32 + c) * 6;
            const float dd = TT[0] * uu0 + TT[1] * uu1 + TT[2] * uu2
                           + 2.f * (TT[3] * uu3 + TT[4] * uu4 + TT[5] * uu5);
            atomicAdd(&agg[rn * 96 + c],       w0 * ss);
            atomicAdd(&agg[rn * 96 + 32 + c],  w1 * dv);
            atomicAdd(&agg[rn * 96 + 64 + c],  w2 * dd);
        }
    }
}

__device__ void edge_pass_V(const P& p, float* agg, const float* u, const float* wp,
                            const float* si, const float* vi, const float* Ti, int m)
{
    for (int e = threadIdx.x; e < N_EDGES; e += 256) {
        const int sn = p.senders[e], rn = p.receivers[e];
        const float ux = u[e * 3], uy = u[e * 3 + 1], uz = u[e * 3 + 2];
        const float um = (m == 0) ? ux : (m == 1) ? uy : uz;
        const float w3 = wp[e * 9 + 3], w4 = wp[e * 9 + 4], w5 = wp[e * 9 + 5];
        for (int c = 0; c < CCH; ++c) {
            const float ss = si[sn * 32 + c];
            const float* vv = vi + (sn * 32 + c) * 3;
            float cr, tu;
            const float* TT = Ti + (sn * 32 + c) * 6;
            if (m == 0)      { cr = vv[1] * uz - vv[2] * uy; tu = TT[0] * ux + TT[3] * uy + TT[4] * uz; }
            else if (m == 1) { cr = vv[2] * ux - vv[0] * uz; tu = TT[3] * ux + TT[1] * uy + TT[5] * uz; }
            else             { cr = vv[0] * uy - vv[1] * ux; tu = TT[4] * ux + TT[5] * uy + TT[2] * uz; }
            atomicAdd(&agg[rn * 96 + c],       w3 * ss * um);
            atomicAdd(&agg[rn * 96 + 32 + c],  w4 * cr);
            atomicAdd(&agg[rn * 96 + 64 + c],  w5 * tu);
        }
    }
}

__device__ void edge_pass_T(const P& p, float* agg, const float* u, const float* wp,
                            const float* si, const float* vi, const float* Ti, int t6)
{
    const int P6[6] = {0, 1, 2, 0, 0, 1};
    const int Q6[6] = {0, 1, 2, 1, 2, 2};
    const int ia = P6[t6], ib = Q6[t6];
    const float third = (ia == ib) ? (1.f / 3.f) : 0.f;
    for (int e = threadIdx.x; e < N_EDGES; e += 256) {
        const int sn = p.senders[e], rn = p.receivers[e];
        const float uv0 = u[e * 3], uv1 = u[e * 3 + 1], uv2 = u[e * 3 + 2];
        const float uvv[3] = {uv0, uv1, uv2};
        const float ua = uvv[ia], ub = uvv[ib];
        const float uupq = ua * ub - third;
        const float w6 = wp[e * 9 + 6], w7 = wp[e * 9 + 7], w8 = wp[e * 9 + 8];
        for (int c = 0; c < CCH; ++c) {
            const float ss = si[sn * 32 + c];
            const float* vv = vi + (sn * 32 + c) * 3;
            const float vdu = vv[0] * uv0 + vv[1] * uv1 + vv[2] * uv2;
            const float sym = 0.5f * (vv[ia] * ub + vv[ib] * ua) - third * vdu;
            const float* TT = Ti + (sn * 32 + c) * 6;
            atomicAdd(&agg[rn * 96 + c],       w6 * ss * uupq);
            atomicAdd(&agg[rn * 96 + 32 + c],  w7 * sym);
            atomicAdd(&agg[rn * 96 + 64 + c],  w8 * TT[t6]);
        }
    }
}

// ---------------------------------------------------------------------------
// Full 3-layer energy, entirely in LDS with one 256-thread (8-wave) block.
// ---------------------------------------------------------------------------
__device__ float energy_eval(const P& p, float* smem, int pi, int pj, float di, float dj)
{
    const int tid = threadIdx.x;
    float* pos   = smem + OFF_POS;
    float* r_e   = smem + OFF_R;
    float* u     = smem + OFF_U;
    float* wpath = smem + OFF_WPATH;
    float* fs[2] = {smem + OFF_S0, smem + OFF_S1};
    float* fv[2] = {smem + OFF_V0, smem + OFF_V1};
    float* fT[2] = {smem + OFF_T0, smem + OFF_T1};
    float* agg   = smem + OFF_AGG;
    float* wc    = smem + OFF_WC;
    float* gv    = smem + OFF_GV;
    float* gT    = smem + OFF_GT;
    float* red   = smem + OFF_RED;

    // perturbed positions
    for (int idx = tid; idx < NDOF; idx += 256) {
        float v = p.positions[idx];
        if (idx == pi) v += di;
        if (idx == pj) v += dj;
        pos[idx] = v;
    }
    // layer-0 features: one-hot species zero-padded to 32 channels; v,T = 0
    for (int idx = tid; idx < 1024; idx += 256) {
        const int n = idx >> 5, c = idx & 31;
        fs[0][idx] = (c == p.species[n]) ? 1.0f : 0.0f;
    }
    for (int idx = tid; idx < 3072; idx += 256) fv[0][idx] = 0.0f;
    for (int idx = tid; idx < 6144; idx += 256) fT[0][idx] = 0.0f;
    __syncthreads();

    // edge geometry
    for (int e = tid; e < N_EDGES; e += 256) {
        const int sn = p.senders[e], rn = p.receivers[e];
        const float rx = (pos[rn * 3]     - pos[sn * 3])     * (1.0f / CUTOFF_R);
        const float ry = (pos[rn * 3 + 1] - pos[sn * 3 + 1]) * (1.0f / CUTOFF_R);
        const float rz = (pos[rn * 3 + 2] - pos[sn * 3 + 2]) * (1.0f / CUTOFF_R);
        const float rr = sqrtf(rx * rx + ry * ry + rz * rz);
        r_e[e] = rr;
        const float inv = 1.0f / (rr + 1e-9f);
        u[e * 3]     = rx * inv;
        u[e * 3 + 1] = ry * inv;
        u[e * 3 + 2] = rz * inv;
    }
    __syncthreads();

    int ping = 0;
    for (int l = 0; l < 3; ++l) {
        float* si = fs[ping];  float* so = fs[ping ^ 1];
        float* vi = fv[ping];  float* vo = fv[ping ^ 1];
        float* Ti = fT[ping];  float* To = fT[ping ^ 1];

        // radial MLP -> 9 path weights per edge
        for (int e = tid; e < N_EDGES; e += 256) {
            const float rr = r_e[e];
            const float rc = fminf(fmaxf(rr, 0.0f), 1.0f);
            const float env = 1.0f + rc * rc * rc * (-10.0f + rc * (15.0f - 6.0f * rc));
            const float inv = 1.0f / (rr + 1e-9f);
            float bas[8];
#pragma unroll
            for (int b = 0; b < 8; ++b)
                bas[b] = sinf(PI_F * (float)(b + 1) * rr) * inv * env;
            float wpl[9] = {0, 0, 0, 0, 0, 0, 0, 0, 0};
            for (int k = 0; k < 64; ++k) {
                float h = 0.0f;
#pragma unroll
                for (int b = 0; b < 8; ++b)
                    h += bas[b] * p.Wr1[(l * 8 + b) * 64 + k];
                h = h * sigmoidf_(h);     // silu
#pragma unroll
                for (int q = 0; q < 9; ++q)
                    wpl[q] += h * p.Wr2[(l * 64 + k) * 9 + q];
            }
#pragma unroll
            for (int q = 0; q < 9; ++q) wpath[e * 9 + q] = wpl[q];
        }
        __syncthreads();

        // ---- scalar irrep ----
        for (int idx = tid; idx < 3072; idx += 256) agg[idx] = 0.0f;
        stage_wcomb(wc, l, 0, p);
        __syncthreads();
        edge_pass_S(p, agg, u, wpath, si, vi, Ti);
        __syncthreads();
        mix_wmma(agg, si, 1, 0, wc, so);
        __syncthreads();
        for (int idx = tid; idx < 1024; idx += 256) {          // silu
            const float x = so[idx];
            so[idx] = x * sigmoidf_(x);
        }
        __syncthreads();
        gates_wmma(so, p.Wg + l * 2 * 32 * 32, gv, gT);
        __syncthreads();
        for (int idx = tid; idx < 2048; idx += 256) {          // sigmoid gates
            float* g = (idx < 1024) ? gv : gT;
            const int k = idx & 1023;
            g[k] = sigmoidf_(g[k]);
        }

        // ---- vector irrep (3 spatial comps) ----
        stage_wcomb(wc, l, 1, p);
        for (int m = 0; m < 3; ++m) {
            for (int idx = tid; idx < 3072; idx += 256) agg[idx] = 0.0f;
            __syncthreads();
            edge_pass_V(p, agg, u, wpath, si, vi, Ti, m);
            __syncthreads();
            mix_wmma(agg, vi, 3, m, wc, vo);
            __syncthreads();
        }
        for (int idx = tid; idx < 3072; idx += 256) vo[idx] *= gv[idx / 3];

        // ---- tensor irrep (6 symmetric comps) ----
        stage_wcomb(wc, l, 2, p);
        for (int t6 = 0; t6 < 6; ++t6) {
            for (int idx = tid; idx < 3072; idx += 256) agg[idx] = 0.0f;
            __syncthreads();
            edge_pass_T(p, agg, u, wpath, si, vi, Ti, t6);
            __syncthreads();
            mix_wmma(agg, Ti, 6, t6, wc, To);
            __syncthreads();
        }
        for (int idx = tid; idx < 6144; idx += 256) To[idx] *= gT[idx / 6];
        __syncthreads();

        ping ^= 1;
    }

    // readout
    const float* sfin = fs[ping];
    if (tid < N_ATOMS) {
        float e = 0.0f;
        for (int k = 0; k < 8; ++k) {
            float t = 0.0f;
            for (int c = 0; c < 32; ++c) t += sfin[tid * 32 + c] * p.Wro1[c * 8 + k];
            e += t * p.Wro2[k];
        }
        red[tid] = e;
    }
    __syncthreads();
    if (tid == 0) {
        float t = 0.0f;
        for (int n = 0; n < N_ATOMS; ++n) t += red[n];
        red[32] = t;
    }
    __syncthreads();
    return red[32];
}

// ---------------------------------------------------------------------------
// One block per upper-triangular (i,j) DOF pair; 4-point central difference.
// ---------------------------------------------------------------------------
__global__ __launch_bounds__(256)
void hess_fd_kernel(P p)
{
    extern __shared__ float smem[];
    if (threadIdx.x == 0) {
        __builtin_prefetch(p.W_s, 0, 1);
        __builtin_prefetch(p.W_v, 0, 1);
        __builtin_prefetch(p.W_T, 0, 1);
        __builtin_prefetch(p.Wg, 0, 1);
    }
    int b = blockIdx.x;
    int i = 0, rem = b;
    while (rem >= (NDOF - i)) { rem -= (NDOF - i); ++i; }
    const int j = i + rem;

    float acc = 0.0f;
    for (int t = 0; t < 4; ++t) {
        const float di = (t & 2) ? -FD_H : FD_H;
        const float dj = (t & 1) ? -FD_H : FD_H;
        const float sgn = ((t == 1) || (t == 2)) ? -1.0f : 1.0f;
        const float E = energy_eval(p, smem, i, j, di, dj);
        acc += sgn * E;
        __syncthreads();
    }
    if (threadIdx.x == 0) {
        const float v = acc / (4.0f * FD_H * FD_H);
        p.out[i * NDOF + j] = v;
        p.out[j * NDOF + i] = v;
    }
}

extern "C" void kernel_launch(void* const* d_in, const int* in_sizes, int n_in,
                              void* d_out, int out_size, void* d_ws, size_t ws_size,
                              hipStream_t stream)
{
    (void)in_sizes; (void)n_in; (void)out_size; (void)d_ws; (void)ws_size;
    P p;
    p.positions = (const float*)d_in[0];
    p.Wr1       = (const float*)d_in[1];
    p.Wr2       = (const float*)d_in[2];
    p.W0_s      = (const float*)d_in[3];
    p.W0_v      = (const float*)d_in[4];
    p.W0_T      = (const float*)d_in[5];
    p.Wsk0_s    = (const float*)d_in[6];
    p.Wsk0_v    = (const float*)d_in[7];
    p.Wsk0_T    = (const float*)d_in[8];
    p.W_s       = (const float*)d_in[9];
    p.W_v       = (const float*)d_in[10];
    p.W_T       = (const float*)d_in[11];
    p.Wsk       = (const float*)d_in[12];
    p.Wg        = (const float*)d_in[13];
    p.Wro1      = (const float*)d_in[14];
    p.Wro2      = (const float*)d_in[15];
    p.species   = (const int*)d_in[16];
    p.senders   = (const int*)d_in[17];
    p.receivers = (const int*)d_in[18];
    p.out       = (float*)d_out;

    const size_t smem_bytes = (size_t)SMEM_FLOATS * sizeof(float);
    hipLaunchKernelGGL(hess_fd_kernel, dim3(NPAIRS), dim3(256), smem_bytes, stream, p);
}